// BuiltCZ_15762529977053
// MI455X (gfx1250) — compile-verified
//
#include <hip/hip_runtime.h>

// CZ gate on a batched state vector — pure bandwidth kernel.
//
// out[b, j] = state[b, j] * diag(j),  diag(j) = -1 iff bits c2 and t2 of j are
// both set (c2 = nq-control-1, t2 = nq-target-1), else +1.
//
// x * (+-1.0f) is exactly a conditional sign-bit flip, so it is done as an
// integer XOR of bit 31 (bit-exact vs the reference). Traffic: 512 MB in +
// 512 MB out => ~46 us floor at 23.3 TB/s. Organized around b128 non-temporal
// loads/stores (no L2 reuse possible: streams exceed the 192 MB L2), 32-bit
// indexing (problem is 2^27 elements) so addressing lowers to saddr+voffset,
// and a wave-uniform fast path that computes one sign per 16B vector when the
// sign bits lie above the low two column bits (true for the given inputs).
// WMMA/TDM do not apply: no matmul structure, zero reuse.

typedef __attribute__((ext_vector_type(4))) unsigned int uint4v;

__global__ __launch_bounds__(256) void cz_sign_flip_kernel(
    const uint4v* __restrict__ in,   // state as 16-byte vectors
    uint4v* __restrict__ out,
    const int* __restrict__ p_control,
    const int* __restrict__ p_target,
    const int* __restrict__ p_num_qubits,
    unsigned n4,                     // number of 16-byte vectors
    unsigned n)                      // total float elements (tail)
{
    // Wave-uniform scalars (lower to s_load).
    const int nq = p_num_qubits[0];
    const int c2 = nq - p_control[0] - 1;
    const int t2 = nq - p_target[0] - 1;
    const unsigned dmask = (nq >= 32) ? 0xFFFFFFFFu : ((1u << nq) - 1u);

    const unsigned gid    = blockIdx.x * blockDim.x + threadIdx.x;
    const unsigned stride = gridDim.x * blockDim.x;

    auto sign_of = [&](unsigned elem_idx) -> unsigned {
        const unsigned col = elem_idx & dmask;
        return (((col >> c2) & (col >> t2)) & 1u) << 31;
    };

    if (c2 >= 2 && t2 >= 2) {
        // Fast path (uniform branch): both sign bits sit above the low two
        // column bits, so all 4 lanes of an aligned vector share one sign.
        unsigned i = gid;
        for (; i + 3u * stride < n4; i += 4u * stride) {
            const unsigned i0 = i;
            const unsigned i1 = i +      stride;
            const unsigned i2 = i + 2u * stride;
            const unsigned i3 = i + 3u * stride;

            uint4v v0 = __builtin_nontemporal_load(&in[i0]);
            uint4v v1 = __builtin_nontemporal_load(&in[i1]);
            uint4v v2 = __builtin_nontemporal_load(&in[i2]);
            uint4v v3 = __builtin_nontemporal_load(&in[i3]);

            const unsigned s0 = sign_of(i0 << 2);
            const unsigned s1 = sign_of(i1 << 2);
            const unsigned s2 = sign_of(i2 << 2);
            const unsigned s3 = sign_of(i3 << 2);

#pragma unroll
            for (int k = 0; k < 4; ++k) {
                v0[k] ^= s0; v1[k] ^= s1; v2[k] ^= s2; v3[k] ^= s3;
            }

            __builtin_nontemporal_store(v0, &out[i0]);
            __builtin_nontemporal_store(v1, &out[i1]);
            __builtin_nontemporal_store(v2, &out[i2]);
            __builtin_nontemporal_store(v3, &out[i3]);
        }
        for (; i < n4; i += stride) {
            uint4v v = __builtin_nontemporal_load(&in[i]);
            const unsigned s = sign_of(i << 2);
#pragma unroll
            for (int k = 0; k < 4; ++k) v[k] ^= s;
            __builtin_nontemporal_store(v, &out[i]);
        }
    } else {
        // General path: per-element sign (c2 or t2 in the low two bits).
        unsigned i = gid;
        for (; i < n4; i += stride) {
            uint4v v = __builtin_nontemporal_load(&in[i]);
            const unsigned b = i << 2;
#pragma unroll
            for (int k = 0; k < 4; ++k) v[k] ^= sign_of(b + (unsigned)k);
            __builtin_nontemporal_store(v, &out[i]);
        }
    }

    // Scalar tail (n not divisible by 4).
    const unsigned tail_base = n4 << 2;
    const unsigned rem       = n - tail_base;
    if (gid < rem) {
        const unsigned* in_s  = (const unsigned*)in;
        unsigned*       out_s = (unsigned*)out;
        const unsigned  e     = tail_base + gid;
        out_s[e] = in_s[e] ^ sign_of(e);
    }
}

extern "C" void kernel_launch(void* const* d_in, const int* in_sizes, int n_in,
                              void* d_out, int out_size, void* d_ws, size_t ws_size,
                              hipStream_t stream) {
    (void)n_in; (void)d_ws; (void)ws_size; (void)out_size;

    const uint4v* state = (const uint4v*)d_in[0];   // float32 [B, 2^nq]
    const int* control  = (const int*)d_in[1];
    const int* target   = (const int*)d_in[2];
    const int* nqubits  = (const int*)d_in[3];
    uint4v* out         = (uint4v*)d_out;

    const unsigned n  = (unsigned)in_sizes[0];
    const unsigned n4 = n >> 2;

    const int threads = 256;
    // One 4-vector main-loop pass per thread (each pass moves 4 x 16 B).
    unsigned want = (n4 + (unsigned)(threads * 4) - 1) / (unsigned)(threads * 4);
    if (want < 1u) want = 1u;
    if (want > 1048576u) want = 1048576u;           // grid-stride covers the rest
    const int blocks = (int)want;

    cz_sign_flip_kernel<<<blocks, threads, 0, stream>>>(
        state, out, control, target, nqubits, n4, n);
}